// DeformableAttention_9371618640484
// MI455X (gfx1250) — compile-verified
//
#include <hip/hip_runtime.h>
#include <hip/hip_bf16.h>

typedef __attribute__((ext_vector_type(2))) float v2f;
typedef __attribute__((ext_vector_type(4))) float v4f;
typedef __attribute__((ext_vector_type(8))) float v8f;

#define NB        2
#define NQ        8192
#define D_MODEL   256
#define KDIM      256
#define N_HEADS   8
#define HEAD_DIM  32
#define N_LEVELS  4
#define N_POINTS  4
#define N_TOK     21760

// ---------------------------------------------------------------------------
// Generic GEMM: out[M,N] = A[M,256] * W[256,N] + bias[N]
// Block = 256 threads = 8 wave32. Each wave computes a 16(M) x 64(N) strip
// as 4 WMMA 16x16 tiles, K-stepped by 4 with V_WMMA_F32_16X16X4_F32.
//
// The 256x64 W strip lives in LDS (64 KB) K-pair-interleaved:
//   element (k, n) at lds[((k>>1)*64 + n)*2 + (k&1)]
// so a B fragment {W[k,n], W[k+1,n]} is one contiguous ds_load_b64.
// Requires: M % 128 == 0, N % 64 == 0 (true for all call sites).
// ---------------------------------------------------------------------------
__launch_bounds__(256)
__global__ void gemm_k256(const float* __restrict__ A, const float* __restrict__ W,
                          const float* __restrict__ bias, float* __restrict__ out,
                          int N) {
  __shared__ float lds[256 * 64];

  const int t    = threadIdx.x;
  const int col0 = blockIdx.y * 64;

  // Cooperative interleaved fill. Unit = (row-pair k2, float4 column j4):
  // 128 row-pairs x 16 float4 cols = 2048 units -> 8 per thread.
#pragma unroll
  for (int i = 0; i < 8; ++i) {
    int u  = t + i * 256;
    int k2 = u >> 4;
    int j4 = u & 15;
    const float* r0 = W + (size_t)(2 * k2) * N + col0 + j4 * 4;
    v4f a = *(const v4f*)r0;        // row 2*k2,   cols j4*4 .. +3
    v4f b = *(const v4f*)(r0 + N);  // row 2*k2+1, cols j4*4 .. +3
    v4f lo = {a.x, b.x, a.y, b.y};
    v4f hi4 = {a.z, b.z, a.w, b.w};
    float* dst = lds + ((size_t)k2 * 64 + j4 * 4) * 2;
    *(v4f*)dst = lo;
    *(v4f*)(dst + 4) = hi4;
  }
  __syncthreads();

  const int wave = t >> 5;
  const int lane = t & 31;
  const int hi   = lane >> 4;   // half-wave select
  const int mn   = lane & 15;   // M (for A) / N (for B) index within tile
  const int row0 = blockIdx.x * 128 + wave * 16;

  const float* Arow = A + (size_t)(row0 + mn) * KDIM;

  v8f acc0 = {}, acc1 = {}, acc2 = {}, acc3 = {};

  for (int kk = 0; kk < KDIM; kk += 64) {
    // Prefetch the next 64-float chunk of this A row (global_prefetch_b8).
    __builtin_prefetch(Arow + kk + 64, 0, 1);
#pragma unroll
    for (int k0 = 0; k0 < 64; k0 += 4) {
      // A fragment 16x4: lane(hi,mn) holds A[mn, k+2*hi], A[mn, k+2*hi+1]
      v2f a = *(const v2f*)(Arow + kk + k0 + 2 * hi);

      // B fragments: one b64 each, {K = k+2*hi, K = k+2*hi+1} at col t*16+mn
      const v2f* bp = (const v2f*)(lds + ((size_t)((kk + k0) / 2 + hi) * 64 + mn) * 2);
      v2f b0 = bp[0];
      v2f b1 = bp[16];
      v2f b2 = bp[32];
      v2f b3 = bp[48];

      acc0 = __builtin_amdgcn_wmma_f32_16x16x4_f32(false, a, false, b0, (short)0, acc0, false, false);
      acc1 = __builtin_amdgcn_wmma_f32_16x16x4_f32(false, a, false, b1, (short)0, acc1, false, false);
      acc2 = __builtin_amdgcn_wmma_f32_16x16x4_f32(false, a, false, b2, (short)0, acc2, false, false);
      acc3 = __builtin_amdgcn_wmma_f32_16x16x4_f32(false, a, false, b3, (short)0, acc3, false, false);
    }
  }

  // Epilogue: C/D layout -> lane holds col n=mn, rows m = r + 8*hi (r = vgpr 0..7)
  v8f accs[4] = {acc0, acc1, acc2, acc3};
#pragma unroll
  for (int tt = 0; tt < 4; ++tt) {
    int n = col0 + tt * 16 + mn;
    float bn = bias[n];
#pragma unroll
    for (int r = 0; r < 8; ++r) {
      int m = row0 + r + 8 * hi;
      out[(size_t)m * N + n] = accs[tt][r] + bn;
    }
  }
}

// ---------------------------------------------------------------------------
// In-place softmax over groups of 4 (points axis). One thread per group.
// attn laid out [B, NQ, H, L, P=4] -> group stride 4 floats (16B aligned).
// ---------------------------------------------------------------------------
__launch_bounds__(256)
__global__ void softmax4(float* __restrict__ attn) {
  size_t g = (size_t)blockIdx.x * blockDim.x + threadIdx.x;
  v4f x = *(v4f*)(attn + g * 4);
  float mx = fmaxf(fmaxf(x.x, x.y), fmaxf(x.z, x.w));
  float e0 = __expf(x.x - mx), e1 = __expf(x.y - mx);
  float e2 = __expf(x.z - mx), e3 = __expf(x.w - mx);
  float inv = 1.0f / (e0 + e1 + e2 + e3);
  v4f y; y.x = e0 * inv; y.y = e1 * inv; y.z = e2 * inv; y.w = e3 * inv;
  *(v4f*)(attn + g * 4) = y;
}

// ---------------------------------------------------------------------------
// Bilinear sampling + attention-weighted accumulation.
// One wave32 per (b, q, h); lane = Dh channel (coalesced 128B corner loads).
// v:    [B, N_TOK, 256]  (channel c = h*32 + dh)
// attn: [B, NQ, H, L, P]
// ref:  [B, NQ, L, 2]    sloc: [H, L, P, 2]
// out:  [B, NQ, 256]
// ---------------------------------------------------------------------------
__launch_bounds__(256)
__global__ void deform_sample(const float* __restrict__ v, const float* __restrict__ attn,
                              const float* __restrict__ ref, const float* __restrict__ sloc,
                              float* __restrict__ out) {
  const int wg   = blockIdx.x * 8 + (threadIdx.x >> 5);
  const int lane = threadIdx.x & 31;
  const int b = wg >> 16;           // wg = (b*NQ + q)*H + h ; NQ*H = 65536
  const int q = (wg >> 3) & (NQ - 1);
  const int h = wg & (N_HEADS - 1);

  const int HL[4] = {128, 64, 32, 16};          // square levels
  const int S0[4] = {0, 16384, 20480, 21504};   // level start indices

  const float* refp = ref  + ((size_t)(b * NQ + q) * N_LEVELS) * 2;
  const float* attp = attn + (((size_t)(b * NQ + q) * N_HEADS + h) * N_LEVELS) * N_POINTS;
  const float* slp  = sloc + ((size_t)h * N_LEVELS * N_POINTS) * 2;
  const float* vb   = v + (size_t)b * N_TOK * D_MODEL + h * HEAD_DIM + lane;

  float acc = 0.0f;
#pragma unroll
  for (int l = 0; l < N_LEVELS; ++l) {
    const int   Wl = HL[l], Hl = HL[l];
    const float fW = (float)Wl, fH = (float)Hl;
    const float* vl = vb + (size_t)S0[l] * D_MODEL;
    const float rx = refp[l * 2 + 0];
    const float ry = refp[l * 2 + 1];
#pragma unroll
    for (int p = 0; p < N_POINTS; ++p) {
      const float aw = attp[l * 4 + p];
      const float lx = rx + slp[(l * 4 + p) * 2 + 0];
      const float ly = ry + slp[(l * 4 + p) * 2 + 1];
      // align_corners=False: ix = ((2*loc - 1 + 1)*W - 1)/2 = loc*W - 0.5
      const float ix = lx * fW - 0.5f;
      const float iy = ly * fH - 0.5f;
      const float x0f = floorf(ix), y0f = floorf(iy);
      const int x0 = (int)x0f, y0 = (int)y0f;
      const int x1 = x0 + 1, y1 = y0 + 1;
      const float wx1 = ix - x0f, wy1 = iy - y0f;
      const float wx0 = 1.0f - wx1, wy0 = 1.0f - wy1;

      const int cx[4] = {x0, x1, x0, x1};
      const int cy[4] = {y0, y0, y1, y1};
      const float cw[4] = {wy0 * wx0, wy0 * wx1, wy1 * wx0, wy1 * wx1};
#pragma unroll
      for (int c = 0; c < 4; ++c) {
        int xi = cx[c], yi = cy[c];
        bool valid = (xi >= 0) && (xi < Wl) && (yi >= 0) && (yi < Hl);
        int xc = xi < 0 ? 0 : (xi > Wl - 1 ? Wl - 1 : xi);
        int yc = yi < 0 ? 0 : (yi > Hl - 1 ? Hl - 1 : yi);
        float val = vl[(size_t)(yc * Wl + xc) * D_MODEL];
        acc += valid ? (aw * cw[c] * val) : 0.0f;
      }
    }
  }
  out[(size_t)(b * NQ + q) * D_MODEL + h * HEAD_DIM + lane] = acc;
}

// ---------------------------------------------------------------------------
extern "C" void kernel_launch(void* const* d_in, const int* in_sizes, int n_in,
                              void* d_out, int out_size, void* d_ws, size_t ws_size,
                              hipStream_t stream) {
  (void)in_sizes; (void)n_in; (void)out_size; (void)ws_size;

  const float* query  = (const float*)d_in[0];   // [2,8192,256]
  const float* refpts = (const float*)d_in[1];   // [2,8192,4,2]
  const float* inflat = (const float*)d_in[2];   // [2,21760,256]
  // d_in[3], d_in[4]: spatial shapes / level starts -> compile-time constants
  const float* Wq = (const float*)d_in[5];
  const float* bq = (const float*)d_in[6];
  const float* Wv = (const float*)d_in[7];
  const float* bv = (const float*)d_in[8];
  const float* Wa = (const float*)d_in[9];       // [256,128]
  const float* ba = (const float*)d_in[10];      // [128]
  const float* sl = (const float*)d_in[11];      // [8,4,4,2]
  const float* Wo = (const float*)d_in[12];
  const float* bo = (const float*)d_in[13];
  float* out = (float*)d_out;

  float* ws = (float*)d_ws;
  float* vproj   = ws;                                       // 43520*256
  float* qproj   = vproj + (size_t)NB * N_TOK * D_MODEL;     // 16384*256
  float* attnbuf = qproj + (size_t)NB * NQ * D_MODEL;        // 16384*128
  float* sampled = attnbuf + (size_t)NB * NQ * (N_HEADS * N_LEVELS * N_POINTS);

  const int NA = N_HEADS * N_LEVELS * N_POINTS;  // 128

  // v = input_flatten @ Wv + bv          M=43520, N=256
  gemm_k256<<<dim3((NB * N_TOK) / 128, D_MODEL / 64), 256, 0, stream>>>(inflat, Wv, bv, vproj, D_MODEL);
  // q = query @ Wq + bq                  M=16384, N=256
  gemm_k256<<<dim3((NB * NQ) / 128, D_MODEL / 64), 256, 0, stream>>>(query, Wq, bq, qproj, D_MODEL);
  // logits = q @ Wa + ba                 M=16384, N=128
  gemm_k256<<<dim3((NB * NQ) / 128, NA / 64), 256, 0, stream>>>(qproj, Wa, ba, attnbuf, NA);
  // softmax over P=4 groups (524288 groups)
  softmax4<<<(NB * NQ * N_HEADS * N_LEVELS) / 256, 256, 0, stream>>>(attnbuf);
  // bilinear sampling + weighted sum     131072 waves
  deform_sample<<<(NB * NQ * N_HEADS) / 8, 256, 0, stream>>>(vproj, attnbuf, refpts, sl, sampled);
  // out = sampled @ Wo + bo              M=16384, N=256
  gemm_k256<<<dim3((NB * NQ) / 128, D_MODEL / 64), 256, 0, stream>>>(sampled, Wo, bo, out, D_MODEL);
}